// TransitionCRF_18236431139185
// MI455X (gfx1250) — compile-verified
//
#include <hip/hip_runtime.h>

// Problem constants (match reference)
#define B_SZ 256
#define T_SZ 2048
#define P_SZ 64
#define TP   (T_SZ * P_SZ)   // 131072

typedef float v2f __attribute__((ext_vector_type(2)));
typedef float v8f __attribute__((ext_vector_type(8)));

// ---------------------------------------------------------------------------
// Kernel 1: CRF forward algorithm (log partition function) via f32 WMMA.
// One wave32 per 16-batch-row tile. 16 blocks x 32 threads total.
//   alpha' = log( exp(alpha - m) @ exp(A) ) + m + u_t
// ---------------------------------------------------------------------------
__global__ __launch_bounds__(32) void crf_forward_kernel(
    const float* __restrict__ unary,   // [B, T, P]
    const float* __restrict__ A,       // [P, P]
    float* __restrict__ logZ)          // [B]
{
    __shared__ float pbuf[16 * 64];    // p tile, row-major [row][64]

    const int lane = threadIdx.x;      // 0..31
    const int l16  = lane & 15;
    const int hi   = lane >> 4;        // 0 = lanes 0-15, 1 = lanes 16-31
    const int b0   = blockIdx.x * 16;  // first batch row of this tile

    // ---- Load B fragments of exp(A), kept in VGPRs for all 2047 steps ----
    // B (4x16) layout per k-iter j, n-chunk c:
    //   V0: lanes0-15 -> K=4j,   N=c*16+lane ; lanes16-31 -> K=4j+2
    //   V1: lanes0-15 -> K=4j+1            ; lanes16-31 -> K=4j+3
    v2f bf[4][16];
#pragma unroll
    for (int c = 0; c < 4; ++c) {
#pragma unroll
        for (int j = 0; j < 16; ++j) {
            const int k0  = 4 * j + 2 * hi;
            const int col = c * 16 + l16;
            bf[c][j].x = __expf(A[(k0    ) * 64 + col]);
            bf[c][j].y = __expf(A[(k0 + 1) * 64 + col]);
        }
    }

    // alpha state in C/D layout: alpha[c][v] = alpha[row = v + 8*hi][col = c*16 + l16]
    // Base pointer for this lane's half-tile of unary.
    const float* ubase = unary + (size_t)(b0 + 8 * hi) * TP + l16;

    float alpha[4][8];
#pragma unroll
    for (int c = 0; c < 4; ++c)
#pragma unroll
        for (int v = 0; v < 8; ++v)
            alpha[c][v] = ubase[v * TP + c * 16];   // t = 0

    float m[8];

    for (int t = 1; t < T_SZ; ++t) {
        // ---- per-row max over all 64 columns (reduce within 16-lane half) ----
#pragma unroll
        for (int v = 0; v < 8; ++v) {
            float mm = fmaxf(fmaxf(alpha[0][v], alpha[1][v]),
                             fmaxf(alpha[2][v], alpha[3][v]));
            mm = fmaxf(mm, __shfl_xor(mm, 1, 32));
            mm = fmaxf(mm, __shfl_xor(mm, 2, 32));
            mm = fmaxf(mm, __shfl_xor(mm, 4, 32));
            mm = fmaxf(mm, __shfl_xor(mm, 8, 32));
            m[v] = mm;
        }

        // ---- p = exp(alpha - m) -> LDS (row-major), for A-fragment transpose ----
#pragma unroll
        for (int c = 0; c < 4; ++c)
#pragma unroll
            for (int v = 0; v < 8; ++v)
                pbuf[(v + 8 * hi) * 64 + c * 16 + l16] =
                    __expf(alpha[c][v] - m[v]);

        __builtin_amdgcn_wave_barrier();   // scheduling fence (DS is in-order in-wave)

        // ---- q = p @ exp(A) : 4 N-chunks x 16 K-iters of V_WMMA_F32_16X16X4_F32 ----
        v8f acc[4];
#pragma unroll
        for (int c = 0; c < 4; ++c)
#pragma unroll
            for (int e = 0; e < 8; ++e)
                acc[c][e] = 0.0f;

#pragma unroll
        for (int j = 0; j < 16; ++j) {
            // A (16x4) fragment: lanes0-15 -> row l16, K=4j..4j+1 ; lanes16-31 -> K=4j+2..4j+3
            const v2f a = *(const v2f*)(&pbuf[l16 * 64 + 4 * j + 2 * hi]);
#pragma unroll
            for (int c = 0; c < 4; ++c)
                acc[c] = __builtin_amdgcn_wmma_f32_16x16x4_f32(
                    false, a, false, bf[c][j], (short)0, acc[c], false, false);
        }

        // ---- alpha' = log(q) + m + u_t  (+ prefetch u_{t+16}) ----
        const float* ut = ubase + t * P_SZ;
#pragma unroll
        for (int c = 0; c < 4; ++c)
#pragma unroll
            for (int v = 0; v < 8; ++v) {
                if (t + 16 < T_SZ)
                    __builtin_prefetch(ut + 16 * P_SZ + v * TP + c * 16, 0, 3);
                const float u = ut[v * TP + c * 16];
                alpha[c][v] = __logf(acc[c][v]) + m[v] + u;
            }
    }

    // ---- logZ[row] = logsumexp over 64 columns ----
#pragma unroll
    for (int v = 0; v < 8; ++v) {
        float mm = fmaxf(fmaxf(alpha[0][v], alpha[1][v]),
                         fmaxf(alpha[2][v], alpha[3][v]));
        mm = fmaxf(mm, __shfl_xor(mm, 1, 32));
        mm = fmaxf(mm, __shfl_xor(mm, 2, 32));
        mm = fmaxf(mm, __shfl_xor(mm, 4, 32));
        mm = fmaxf(mm, __shfl_xor(mm, 8, 32));
        float s = __expf(alpha[0][v] - mm) + __expf(alpha[1][v] - mm) +
                  __expf(alpha[2][v] - mm) + __expf(alpha[3][v] - mm);
        s += __shfl_xor(s, 1, 32);
        s += __shfl_xor(s, 2, 32);
        s += __shfl_xor(s, 4, 32);
        s += __shfl_xor(s, 8, 32);
        if (l16 == 0)
            logZ[b0 + v + 8 * hi] = mm + __logf(s);
    }
}

// ---------------------------------------------------------------------------
// Kernel 2: gold-path score (emission + transition) and per-batch NLL.
// One block per batch element.
// ---------------------------------------------------------------------------
__global__ __launch_bounds__(256) void crf_score_kernel(
    const float* __restrict__ unary,
    const int*   __restrict__ labels,  // [B, T]
    const float* __restrict__ A,
    const float* __restrict__ logZ,
    float* __restrict__ nll)           // [B]
{
    const int b   = blockIdx.x;
    const int tid = threadIdx.x;
    const int*   yb = labels + b * T_SZ;
    const float* ub = unary + (size_t)b * TP;

    float acc = 0.0f;
    for (int t = tid; t < T_SZ; t += 256) {
        const int y = yb[t];
        float v = ub[t * P_SZ + y];               // emission
        if (t >= 1)
            v += A[yb[t - 1] * P_SZ + y];         // transition
        acc += v;
    }

    __shared__ float red[256];
    red[tid] = acc;
    __syncthreads();
#pragma unroll
    for (int s = 128; s > 0; s >>= 1) {
        if (tid < s) red[tid] += red[tid + s];
        __syncthreads();
    }
    if (tid == 0)
        nll[b] = logZ[b] - red[0];
}

// ---------------------------------------------------------------------------
// Kernel 3: mean over batch -> scalar output
// ---------------------------------------------------------------------------
__global__ __launch_bounds__(256) void crf_mean_kernel(
    const float* __restrict__ nll, float* __restrict__ out)
{
    const int tid = threadIdx.x;
    __shared__ float red[256];
    red[tid] = nll[tid];
    __syncthreads();
#pragma unroll
    for (int s = 128; s > 0; s >>= 1) {
        if (tid < s) red[tid] += red[tid + s];
        __syncthreads();
    }
    if (tid == 0)
        out[0] = red[0] * (1.0f / (float)B_SZ);
}

// ---------------------------------------------------------------------------
extern "C" void kernel_launch(void* const* d_in, const int* in_sizes, int n_in,
                              void* d_out, int out_size, void* d_ws, size_t ws_size,
                              hipStream_t stream)
{
    const float* unary  = (const float*)d_in[0];   // [256, 2048, 64] f32
    const int*   labels = (const int*)d_in[1];     // [256, 2048] int
    const float* A      = (const float*)d_in[2];   // [64, 64] f32

    float* logZ = (float*)d_ws;          // 256 floats
    float* nll  = logZ + B_SZ;           // 256 floats
    float* out  = (float*)d_out;         // scalar f32

    crf_forward_kernel<<<dim3(B_SZ / 16), dim3(32), 0, stream>>>(unary, A, logZ);
    crf_score_kernel<<<dim3(B_SZ), dim3(256), 0, stream>>>(unary, labels, A, logZ, nll);
    crf_mean_kernel<<<dim3(1), dim3(256), 0, stream>>>(nll, out);
}